// InitialEdgeDecoder_14620068675786
// MI455X (gfx1250) — compile-verified
//
#include <hip/hip_runtime.h>
#include <math.h>

#define AS3 __attribute__((address_space(3)))

#define B_GRAPHS 4096
#define D_INLAT  256
#define H1DIM    512
#define H2DIM    1024
#define MAX_E    128
#define MAX_N    64
#define NEF      16
#define EDGE_N   (MAX_E * MAX_N)        /* 8192 */
#define FEAT_N   (MAX_E * (NEF - 2))    /* 1792 */

#define KC           16                  /* K-chunk rows staged in LDS      */
#define NBLK         256                 /* output columns per block        */
#define CHUNK_FLOATS (KC * NBLK)         /* 4096 floats = 16KB per buffer   */

typedef float v2f __attribute__((ext_vector_type(2)));
typedef float v4f __attribute__((ext_vector_type(4)));
typedef float v8f __attribute__((ext_vector_type(8)));
typedef int   gv4i __attribute__((vector_size(16)));   /* matches builtin param */

#if __has_builtin(__builtin_amdgcn_global_load_async_to_lds_b128) && \
    __has_builtin(__builtin_amdgcn_s_wait_asynccnt)
#define USE_ASYNC_COPY 1
#else
#define USE_ASYNC_COPY 0
#endif

// ---------------------------------------------------------------------------
// Issue the global->LDS copy of one KC x NBLK chunk of W (row k0 window,
// column window col_blk..col_blk+255), 128 threads, b128 granules.
// ---------------------------------------------------------------------------
__device__ __forceinline__
void issue_chunk(const float* __restrict__ base /* = W + k0*N + col_blk */,
                 int N, AS3 float* dst, int tid
#if !USE_ASYNC_COPY
                 , v4f* stage
#endif
                 )
{
#pragma unroll
    for (int rd = 0; rd < 8; ++rd) {
        int i = (rd * 128 + tid) * 4;     // flat float index in chunk
        int r = i >> 8;                   // row within chunk  (i / 256)
        int c = i & 255;                  // col within chunk
        const float* g = base + (size_t)r * N + c;
#if USE_ASYNC_COPY
        __builtin_amdgcn_global_load_async_to_lds_b128(
            (gv4i*)g, (AS3 gv4i*)(dst + i), 0, 0);
#else
        stage[rd] = *(const v4f*)g;
#endif
    }
}

#if !USE_ASYNC_COPY
__device__ __forceinline__
void commit_chunk(AS3 float* dst, int tid, const v4f* stage)
{
#pragma unroll
    for (int rd = 0; rd < 8; ++rd) {
        int i = (rd * 128 + tid) * 4;
        *(AS3 v4f*)(dst + i) = stage[rd];
    }
}
#endif

// ---------------------------------------------------------------------------
// Double-buffered WMMA MAC loop over K. Each of 4 waves owns a 16x64 tile of
// the block's 16x256 window. B fragments come from LDS (ds_read2st64 pairs),
// A fragments (v2f per lane) straight from L2.
//
// Fragment layouts per CDNA5 ISA 7.12.2 (wave32):
//   A 16x4:  lane l holds A[l&15][k0 + 2*(l>>4) + {0,1}]
//   B 4x16:  lane l holds W[k0 + 2*(l>>4) + {0,1}][n = l&15]
//   C/D:     vgpr v, lane l -> row (v + 8*(l>>4)), col (l&15)
// ---------------------------------------------------------------------------
__device__ __forceinline__
void mac_block(const float* __restrict__ A, const float* __restrict__ W,
               int K, int N, int row_base, int col_blk,
               AS3 float* lds, v8f acc[4])
{
    const int tid   = threadIdx.x;
    const int lane  = tid & 31;
    const int wave  = tid >> 5;
    const int half  = lane >> 4;
    const int l15   = lane & 15;
    const int nbase = wave * 64 + l15;

    const float* Arow = A + (size_t)(row_base + l15) * K + 2 * half;
    acc[0] = {}; acc[1] = {}; acc[2] = {}; acc[3] = {};

    const int nchunks = K / KC;

#if !USE_ASYNC_COPY
    v4f stage[8];
#endif

    // prime chunk 0
#if USE_ASYNC_COPY
    issue_chunk(W + col_blk, N, lds, tid);
    __builtin_amdgcn_s_wait_asynccnt(0);
#else
    issue_chunk(W + col_blk, N, lds, tid, stage);
    commit_chunk(lds, tid, stage);
#endif
    __syncthreads();

    for (int kc = 0; kc < nchunks; ++kc) {
        AS3 float* bufc = lds + ((kc)     & 1) * CHUNK_FLOATS;
        AS3 float* bufn = lds + ((kc + 1) & 1) * CHUNK_FLOATS;
        const int k0 = kc * KC;

        if (kc + 1 < nchunks) {
#if USE_ASYNC_COPY
            issue_chunk(W + (size_t)(k0 + KC) * N + col_blk, N, bufn, tid);
#else
            issue_chunk(W + (size_t)(k0 + KC) * N + col_blk, N, bufn, tid, stage);
#endif
        }

        // ---- compute this chunk: 4 K-steps x 4 N-tiles -------------------
        const AS3 float* Bb = bufc + 2 * half * NBLK + nbase;
#pragma unroll
        for (int kk = 0; kk < 4; ++kk) {
            v2f a = *(const v2f*)(Arow + k0 + kk * 4);
            const AS3 float* Br = Bb + kk * 4 * NBLK;
            v2f b0, b1, b2, b3;
            b0.x = Br[0];       b0.y = Br[NBLK];          // ds_read2st64 pair
            b1.x = Br[16];      b1.y = Br[NBLK + 16];
            b2.x = Br[32];      b2.y = Br[NBLK + 32];
            b3.x = Br[48];      b3.y = Br[NBLK + 48];
            acc[0] = __builtin_amdgcn_wmma_f32_16x16x4_f32(false, a, false, b0, (short)0, acc[0], false, false);
            acc[1] = __builtin_amdgcn_wmma_f32_16x16x4_f32(false, a, false, b1, (short)0, acc[1], false, false);
            acc[2] = __builtin_amdgcn_wmma_f32_16x16x4_f32(false, a, false, b2, (short)0, acc[2], false, false);
            acc[3] = __builtin_amdgcn_wmma_f32_16x16x4_f32(false, a, false, b3, (short)0, acc[3], false, false);
        }

        if (kc + 1 < nchunks) {
#if USE_ASYNC_COPY
            __builtin_amdgcn_s_wait_asynccnt(0);   // own share of chunk kc+1 landed
#else
            commit_chunk(bufn, tid, stage);
#endif
        }
        __syncthreads();                            // chunk kc+1 visible, bufc reusable
    }
}

// ---------------------------------------------------------------------------
// C = act(A @ W + bias). grid = (rows/16, N/256), block = 128.
// ---------------------------------------------------------------------------
__global__ __launch_bounds__(128)
void gemm_bias_act(const float* __restrict__ A, const float* __restrict__ W,
                   const float* __restrict__ bias, float* __restrict__ Out,
                   int K, int N, int do_relu)
{
    __shared__ __align__(16) float smem[2 * CHUNK_FLOATS];
    const int lane = threadIdx.x & 31;
    const int wave = threadIdx.x >> 5;
    const int half = lane >> 4;
    const int l15  = lane & 15;
    const int row_base = blockIdx.x << 4;
    const int col_blk  = blockIdx.y * NBLK;

    v8f acc[4];
    mac_block(A, W, K, N, row_base, col_blk, (AS3 float*)smem, acc);

    const int c0 = col_blk + wave * 64 + l15;
#pragma unroll
    for (int v = 0; v < 8; ++v) {
        int r = row_base + v + 8 * half;
        float* orow = Out + (size_t)r * N;
        float x0 = acc[0][v] + bias[c0];
        float x1 = acc[1][v] + bias[c0 + 16];
        float x2 = acc[2][v] + bias[c0 + 32];
        float x3 = acc[3][v] + bias[c0 + 48];
        if (do_relu) {
            x0 = fmaxf(x0, 0.f); x1 = fmaxf(x1, 0.f);
            x2 = fmaxf(x2, 0.f); x3 = fmaxf(x3, 0.f);
        }
        orow[c0]      = x0;
        orow[c0 + 16] = x1;
        orow[c0 + 32] = x2;
        orow[c0 + 48] = x3;
    }
}

// ---------------------------------------------------------------------------
// Fused edge layer-3 + softmax + top-2. Each wave = one edge's 64 node cols,
// block covers 4 consecutive edges = 256 cols. grid = (B/16, 128/4).
// ---------------------------------------------------------------------------
__global__ __launch_bounds__(128)
void edge_head(const float* __restrict__ H, const float* __restrict__ W3,
               const float* __restrict__ b3, float* __restrict__ senders,
               float* __restrict__ receivers, float* __restrict__ features)
{
    __shared__ __align__(16) float smem[2 * CHUNK_FLOATS];
    __shared__ float lg[4][16][65];     // +1 pad against bank conflicts

    const int lane = threadIdx.x & 31;
    const int wave = threadIdx.x >> 5;
    const int half = lane >> 4;
    const int l15  = lane & 15;
    const int row_base = blockIdx.x << 4;
    const int edge     = blockIdx.y * 4 + wave;
    const int col_blk  = blockIdx.y * NBLK;        // == (edge/4)*256

    v8f acc[4];
    mac_block(H, W3, H2DIM, EDGE_N, row_base, col_blk, (AS3 float*)smem, acc);

    const int cb = edge * MAX_N;                   // bias base for this edge
#pragma unroll
    for (int v = 0; v < 8; ++v) {
        int r = v + 8 * half;
        lg[wave][r][l15]      = acc[0][v] + b3[cb + l15];
        lg[wave][r][l15 + 16] = acc[1][v] + b3[cb + l15 + 16];
        lg[wave][r][l15 + 32] = acc[2][v] + b3[cb + l15 + 32];
        lg[wave][r][l15 + 48] = acc[3][v] + b3[cb + l15 + 48];
    }
    __syncthreads();

    if (lane < 16) {
        const float* row = lg[wave][lane];
        // top-2 with lowest-index-first tie-break, matching lax.top_k
        float v0 = -INFINITY, v1 = -INFINITY;
        int   i0 = 0, i1 = 0;
#pragma unroll 4
        for (int j = 0; j < MAX_N; ++j) {
            float x = row[j];
            if (x > v0)      { v1 = v0; i1 = i0; v0 = x; i0 = j; }
            else if (x > v1) { v1 = x;  i1 = j; }
        }
        float s = 0.f;
#pragma unroll 4
        for (int j = 0; j < MAX_N; ++j) s += expf(row[j] - v0);
        float p0 = 1.0f / s;            // exp(0)/s
        float p1 = expf(v1 - v0) / s;

        int b = row_base + lane;
        size_t be = (size_t)b * MAX_E + edge;
        senders[be]   = (float)(i0 + b * MAX_N);
        receivers[be] = (float)(i1 + b * MAX_N);
        features[be * NEF + 0] = p0;
        features[be * NEF + 1] = p1;
    }
}

// ---------------------------------------------------------------------------
// Fused feature layer-3: out col c -> features[b][e=c/14][2 + c%14].
// grid = (B/16, 1792/256 = 7).
// ---------------------------------------------------------------------------
__global__ __launch_bounds__(128)
void feature_head(const float* __restrict__ H, const float* __restrict__ W3,
                  const float* __restrict__ b3, float* __restrict__ features)
{
    __shared__ __align__(16) float smem[2 * CHUNK_FLOATS];
    const int lane = threadIdx.x & 31;
    const int wave = threadIdx.x >> 5;
    const int half = lane >> 4;
    const int l15  = lane & 15;
    const int row_base = blockIdx.x << 4;
    const int col_blk  = blockIdx.y * NBLK;

    v8f acc[4];
    mac_block(H, W3, H2DIM, FEAT_N, row_base, col_blk, (AS3 float*)smem, acc);

#pragma unroll
    for (int v = 0; v < 8; ++v) {
        int r = row_base + v + 8 * half;
        size_t rbase = (size_t)r * MAX_E * NEF;
#pragma unroll
        for (int t = 0; t < 4; ++t) {
            int c = col_blk + wave * 64 + t * 16 + l15;
            float val = acc[t][v] + b3[c];
            int e = c / (NEF - 2);
            int j = c - e * (NEF - 2);
            features[rbase + (size_t)e * NEF + 2 + j] = val;
        }
    }
}

// ---------------------------------------------------------------------------
extern "C" void kernel_launch(void* const* d_in, const int* in_sizes, int n_in,
                              void* d_out, int out_size, void* d_ws, size_t ws_size,
                              hipStream_t stream)
{
    const float* x   = (const float*)d_in[0];
    const float* eW1 = (const float*)d_in[1];
    const float* eb1 = (const float*)d_in[2];
    const float* eW2 = (const float*)d_in[3];
    const float* eb2 = (const float*)d_in[4];
    const float* eW3 = (const float*)d_in[5];
    const float* eb3 = (const float*)d_in[6];
    const float* fW1 = (const float*)d_in[7];
    const float* fb1 = (const float*)d_in[8];
    const float* fW2 = (const float*)d_in[9];
    const float* fb2 = (const float*)d_in[10];
    const float* fW3 = (const float*)d_in[11];
    const float* fb3 = (const float*)d_in[12];

    float* senders   = (float*)d_out;
    float* receivers = senders + (size_t)B_GRAPHS * MAX_E;
    float* features  = receivers + (size_t)B_GRAPHS * MAX_E;

    float* h1e = (float*)d_ws;
    float* h2e = h1e + (size_t)B_GRAPHS * H1DIM;
    float* h1f = h2e + (size_t)B_GRAPHS * H2DIM;
    float* h2f = h1f + (size_t)B_GRAPHS * H1DIM;

    dim3 blk(128);

    // edge MLP trunk
    gemm_bias_act<<<dim3(B_GRAPHS / 16, H1DIM / NBLK), blk, 0, stream>>>(x,   eW1, eb1, h1e, D_INLAT, H1DIM, 1);
    gemm_bias_act<<<dim3(B_GRAPHS / 16, H2DIM / NBLK), blk, 0, stream>>>(h1e, eW2, eb2, h2e, H1DIM,  H2DIM, 1);
    // fused layer3 + softmax + top2
    edge_head<<<dim3(B_GRAPHS / 16, MAX_E / 4), blk, 0, stream>>>(h2e, eW3, eb3, senders, receivers, features);

    // feature MLP trunk
    gemm_bias_act<<<dim3(B_GRAPHS / 16, H1DIM / NBLK), blk, 0, stream>>>(x,   fW1, fb1, h1f, D_INLAT, H1DIM, 1);
    gemm_bias_act<<<dim3(B_GRAPHS / 16, H2DIM / NBLK), blk, 0, stream>>>(h1f, fW2, fb2, h2f, H1DIM,  H2DIM, 1);
    // fused layer3 with interleaved-store remap
    feature_head<<<dim3(B_GRAPHS / 16, FEAT_N / NBLK), blk, 0, stream>>>(h2f, fW3, fb3, features);
}